// PathWeightedLatticeCore_46291157516409
// MI455X (gfx1250) — compile-verified
//
#include <hip/hip_runtime.h>
#include <math.h>

// ---------------------------------------------------------------------------
// Problem constants (fixed by the reference: B=8, S=8192, D=1024)
// ---------------------------------------------------------------------------
#define BB 8
#define SS 8192
#define DD 1024

typedef float v2f __attribute__((ext_vector_type(2)));
typedef float v8f __attribute__((ext_vector_type(8)));
typedef float f4  __attribute__((ext_vector_type(4)));

// D = A(16x4 f32) * B(4x16 f32) + C(16x16 f32), wave32 WMMA
__device__ __forceinline__ v8f wmma4(v2f a, v2f b, v8f c) {
  return __builtin_amdgcn_wmma_f32_16x16x4_f32(false, a, false, b, (short)0, c,
                                               false, false);
}

__device__ __forceinline__ float sigmoidf_(float x) {
  return 1.0f / (1.0f + expf(-x));
}
__device__ __forceinline__ float gelu_exact(float x) {
  return 0.5f * x * (1.0f + erff(x * 0.70710678118654752440f));
}
__device__ __forceinline__ float softplusf_(float x) {
  return (x > 20.0f) ? x : log1pf(expf(x));
}

// ---------------------------------------------------------------------------
// Host-side lattice analysis table (replicates the Python exactly, including
// the levels[lvl+1]-overwrite behavior and pc.get(a, 1) default).
// ---------------------------------------------------------------------------
#define NSMAX 8
#define AMAX 16

struct StructTable {
  int nstruct;
  int npairs;
  int pos[NSMAX];
  int A[NSMAX];
  int pairBase[NSMAX + 1];
  int anc[NSMAX][AMAX];
  float cnt[NSMAX][AMAX];
};

static int spine_index(const int* spine, int ns, int v) {
  for (int i = 0; i < ns; ++i)
    if (spine[i] == v) return i;
  return -1;
}

static void build_table(StructTable& T) {
  int spine[24];
  int ns = 0;
  spine[ns++] = 0; spine[ns++] = 2; spine[ns++] = 4;
  for (;;) {
    long nxt = 2L * (spine[ns - 1] + spine[ns - 2] + spine[ns - 3]);
    if (nxt >= SS) break;
    spine[ns++] = (int)nxt;
  }
  T.nstruct = 0;
  T.pairBase[0] = 0;
  for (int si = 0; si < ns; ++si) {
    int pos = spine[si];
    if (pos >= SS || pos < 3) continue;
    // ---- BFS (_analyze) ----
    int visited[64]; int nvis = 0; visited[nvis++] = pos;
    int levNode[16][8]; int levCnt[16];
    for (int i = 0; i < 16; ++i) levCnt[i] = 0;
    levNode[0][0] = pos; levCnt[0] = 1;
    int qn[64], ql[64]; int qh = 0, qt = 0;
    qn[qt] = pos; ql[qt] = 0; qt++;
    int maxLvl = 0;
    while (qh < qt) {
      int cur = qn[qh], lvl = ql[qh]; qh++;
      if (lvl >= 9) break;
      int ci = spine_index(spine, ns, cur);
      int ia[3]; int nia = 0;
      if (ci >= 3) { ia[nia++] = spine[ci - 1]; ia[nia++] = spine[ci - 2]; ia[nia++] = spine[ci - 3]; }
      int newl[3]; int nnew = 0;
      for (int j = 0; j < nia; ++j) {
        int a = ia[j];
        bool vis = false;
        for (int v = 0; v < nvis; ++v) if (visited[v] == a) { vis = true; break; }
        if (!vis && a >= 0) { visited[nvis++] = a; newl[nnew++] = a; qn[qt] = a; ql[qt] = lvl + 1; qt++; }
      }
      if (nnew) {  // NOTE: assignment (overwrite), exactly like the Python
        levCnt[lvl + 1] = nnew;
        for (int j = 0; j < nnew; ++j) levNode[lvl + 1][j] = newl[j];
        if (lvl + 1 > maxLvl) maxLvl = lvl + 1;
      }
    }
    // ---- path counts ----
    int pcNode[64]; float pcVal[64]; int npc = 0;
    for (int lvl = maxLvl; lvl >= 0; --lvl) {
      for (int j = 0; j < levCnt[lvl]; ++j) {
        int node = levNode[lvl][j];
        if (node == pos) continue;
        if (lvl == maxLvl) { pcNode[npc] = node; pcVal[npc] = 1.0f; npc++; continue; }
        float c = 0.0f;
        for (int cj = 0; cj < levCnt[lvl + 1]; ++cj) {
          int ch = levNode[lvl + 1][cj];
          int chi = spine_index(spine, ns, ch);
          bool isanc = false;
          if (chi >= 3)
            for (int q = 1; q <= 3; ++q)
              if (spine[chi - q] == node) isanc = true;
          if (isanc) {
            float pv = 0.0f;
            for (int v = 0; v < npc; ++v) if (pcNode[v] == ch) { pv = pcVal[v]; break; }
            c += pv;
          }
        }
        if (lvl != 0) { pcNode[npc] = node; pcVal[npc] = c; npc++; }
      }
    }
    // ---- collect anc / cnt ----
    int A = 0;
    for (int lvl = 1; lvl <= maxLvl; ++lvl) {
      for (int j = 0; j < levCnt[lvl]; ++j) {
        int a = levNode[lvl][j];
        if (a < SS && A < AMAX) {
          T.anc[T.nstruct][A] = a;
          float pv = 1.0f;  // pc.get(a, 1)
          for (int v = 0; v < npc; ++v) if (pcNode[v] == a) { pv = pcVal[v]; break; }
          T.cnt[T.nstruct][A] = pv;
          A++;
        }
      }
    }
    if (A > 0) {
      T.pos[T.nstruct] = pos;
      T.A[T.nstruct] = A;
      T.pairBase[T.nstruct + 1] = T.pairBase[T.nstruct] + A;
      T.nstruct++;
    }
  }
  T.npairs = T.pairBase[T.nstruct];
}

// ---------------------------------------------------------------------------
// Kernel 1: bulk copy x -> out (memory bound: 512 MB, the dominant cost).
// Non-temporal B128 path (x is 256 MB > 192 MB L2, caching it is useless).
// ---------------------------------------------------------------------------
__global__ __launch_bounds__(256) void copy_kernel(const f4* __restrict__ src,
                                                   f4* __restrict__ dst, int n) {
  int i = blockIdx.x * 256 + threadIdx.x;
  if (i < n) {
    f4 v = __builtin_nontemporal_load(src + i);
    __builtin_nontemporal_store(v, dst + i);
  }
}

// ---------------------------------------------------------------------------
// Kernel 2: per (structure, ancestor) pair:
//   w = softplus(relu(cnt*pw_w1+pw_b1)@pw_w2 + pw_b2)
//   m = [h_anc, h_cur] @ msg_w.T + msg_b      (WMMA f32, M=16 pad, N=1024, K=2048)
//   m = gelu(LN(m)) * w   -> mbuf[pair][b][d]
// Block: 256 threads = 8 waves, each wave does 8 N-tiles of 16.
// ---------------------------------------------------------------------------
__global__ __launch_bounds__(256) void msg_kernel(
    const float* __restrict__ x, const float* __restrict__ pw_w1,
    const float* __restrict__ pw_b1, const float* __restrict__ pw_w2,
    const float* __restrict__ pw_b2, const float* __restrict__ msg_w,
    const float* __restrict__ msg_b, const float* __restrict__ ln_g,
    const float* __restrict__ ln_b, float* __restrict__ mbuf, StructTable T) {
  __shared__ float mt[BB][DD];   // 32 KB
  __shared__ float red[64];
  __shared__ float wsh;

  int p = blockIdx.x;
  // locate (structure s, ancestor index t)
  int s = 0;
  while (T.pairBase[s + 1] <= p) s++;
  int t = p - T.pairBase[s];
  int pos = T.pos[s];
  int anc = T.anc[s][t];

  int tid = threadIdx.x;
  // ---- path-weight MLP (tiny) ----
  float cnt = T.cnt[s][t];
  if (tid < 64)
    red[tid] = fmaxf(cnt * pw_w1[tid] + pw_b1[tid], 0.0f) * pw_w2[tid];
  __syncthreads();
  if (tid == 0) {
    float acc = pw_b2[0];
    for (int i = 0; i < 64; ++i) acc += red[i];
    wsh = softplusf_(acc);
  }
  __syncthreads();
  float wval = wsh;

  // ---- msg GEMM via WMMA f32 16x16x4 ----
  int wid = tid >> 5;
  int lane = tid & 31;
  int half = lane >> 4;
  int lcol = lane & 15;
  int mrow = (lcol < 8) ? lcol : 7;
  float mmask = (lcol < 8) ? 1.0f : 0.0f;
  const float* arow0 = x + ((size_t)mrow * SS + anc) * DD;  // k in [0,1024)
  const float* arow1 = x + ((size_t)mrow * SS + pos) * DD;  // k in [1024,2048)

  for (int tile = wid; tile < DD / 16; tile += 8) {
    int n0 = tile * 16;
    int nc = n0 + lcol;
    const float* wrow = msg_w + (size_t)nc * (2 * DD);  // B[k][nc] = msg_w[nc*2D+k]
    v8f acc = {0.f, 0.f, 0.f, 0.f, 0.f, 0.f, 0.f, 0.f};
#pragma unroll 4
    for (int kk = 0; kk < (2 * DD) / 4; ++kk) {
      int k0 = kk * 4 + 2 * half;
      const float* ap = (k0 < DD) ? (arow0 + k0) : (arow1 + (k0 - DD));
      v2f a, b;
      a[0] = ap[0] * mmask;
      a[1] = ap[1] * mmask;
      b[0] = wrow[k0];
      b[1] = wrow[k0 + 1];
      acc = wmma4(a, b, acc);
    }
    if (half == 0) {
      float bias = msg_b[nc];
#pragma unroll
      for (int r = 0; r < 8; ++r) mt[r][nc] = acc[r] + bias;
    }
  }
  __syncthreads();

  // ---- LayerNorm + GELU + scale; wave `wid` owns batch row `wid` ----
  int row = wid;
  float s1 = 0.0f, s2 = 0.0f;
  for (int d = lane; d < DD; d += 32) {
    float v = mt[row][d];
    s1 += v;
    s2 += v * v;
  }
  for (int off = 16; off > 0; off >>= 1) {
    s1 += __shfl_down(s1, off);
    s2 += __shfl_down(s2, off);
  }
  s1 = __shfl(s1, 0);
  s2 = __shfl(s2, 0);
  float mu = s1 * (1.0f / DD);
  float var = s2 * (1.0f / DD) - mu * mu;
  float inv = rsqrtf(var + 1e-5f);
  float* orow = mbuf + ((size_t)p * BB + row) * DD;
  for (int d = lane; d < DD; d += 32) {
    float v = (mt[row][d] - mu) * inv * ln_g[d] + ln_b[d];
    orow[d] = gelu_exact(v) * wval;
  }
}

// ---------------------------------------------------------------------------
// Kernel 3: gi = m @ w_ih.T + b_ih  for all pairs in parallel.
// Grid: (npairs, 24); block 256 threads = 8 waves, one 16-wide N-tile each.
// ---------------------------------------------------------------------------
__global__ __launch_bounds__(256) void gi_kernel(
    const float* __restrict__ mbuf, const float* __restrict__ w_ih,
    const float* __restrict__ b_ih, float* __restrict__ gibuf) {
  int p = blockIdx.x;
  int tid = threadIdx.x;
  int wid = tid >> 5;
  int lane = tid & 31;
  int half = lane >> 4;
  int lcol = lane & 15;
  int tile = blockIdx.y * 8 + wid;  // 0..191
  int n0 = tile * 16;
  int nc = n0 + lcol;

  int mrow = (lcol < 8) ? lcol : 7;
  float mmask = (lcol < 8) ? 1.0f : 0.0f;
  const float* arow = mbuf + ((size_t)p * BB + mrow) * DD;
  const float* wrow = w_ih + (size_t)nc * DD;  // B[k][nc] = w_ih[nc*D+k]

  v8f acc = {0.f, 0.f, 0.f, 0.f, 0.f, 0.f, 0.f, 0.f};
#pragma unroll 4
  for (int kk = 0; kk < DD / 4; ++kk) {
    int k0 = kk * 4 + 2 * half;
    v2f a, b;
    a[0] = arow[k0] * mmask;
    a[1] = arow[k0 + 1] * mmask;
    b[0] = wrow[k0];
    b[1] = wrow[k0 + 1];
    acc = wmma4(a, b, acc);
  }
  if (half == 0) {
    float bias = b_ih[nc];
#pragma unroll
    for (int r = 0; r < 8; ++r)
      gibuf[((size_t)p * BB + r) * (3 * DD) + nc] = acc[r] + bias;
  }
}

// ---------------------------------------------------------------------------
// Kernel 4: per structure — sequential GRU scan over ancestors, then gate
// GEMM and final blend into out[:, pos, :].
// One block of 1024 threads (32 waves) per structure; h lives in LDS (32 KB),
// gh round-trips through workspace (fits easily in L2).
// ---------------------------------------------------------------------------
__global__ __launch_bounds__(1024) void gru_gate_kernel(
    const float* __restrict__ x, const float* __restrict__ gibuf,
    const float* __restrict__ w_hh, const float* __restrict__ b_hh,
    const float* __restrict__ gate_w, const float* __restrict__ gate_b,
    float* __restrict__ ghbuf, float* __restrict__ out, StructTable T) {
  __shared__ float h[BB][DD];  // 32 KB
  int s = blockIdx.x;
  int A = T.A[s];
  int pos = T.pos[s];
  int pbase = T.pairBase[s];

  int tid = threadIdx.x;
  for (int i = tid; i < BB * DD; i += 1024) ((float*)h)[i] = 0.0f;
  __syncthreads();

  int wid = tid >> 5;
  int lane = tid & 31;
  int half = lane >> 4;
  int lcol = lane & 15;
  int mrow = (lcol < 8) ? lcol : 7;
  float mmask = (lcol < 8) ? 1.0f : 0.0f;
  float* gh = ghbuf + (size_t)s * BB * (3 * DD);

  for (int t = 0; t < A; ++t) {
    // gh = h @ w_hh.T + b_hh : M=16(pad), N=3072, K=1024; 192 tiles / 32 waves
    for (int tile = wid; tile < 192; tile += 32) {
      int nc = tile * 16 + lcol;
      const float* wrow = w_hh + (size_t)nc * DD;
      v8f acc = {0.f, 0.f, 0.f, 0.f, 0.f, 0.f, 0.f, 0.f};
#pragma unroll 4
      for (int kk = 0; kk < DD / 4; ++kk) {
        int k0 = kk * 4 + 2 * half;
        v2f a, b;
        a[0] = h[mrow][k0] * mmask;
        a[1] = h[mrow][k0 + 1] * mmask;
        b[0] = wrow[k0];
        b[1] = wrow[k0 + 1];
        acc = wmma4(a, b, acc);
      }
      if (half == 0) {
        float bias = b_hh[nc];
#pragma unroll
        for (int r = 0; r < 8; ++r) gh[(size_t)r * (3 * DD) + nc] = acc[r] + bias;
      }
    }
    __threadfence();
    __syncthreads();
    // element-wise gates: thread owns column d = tid for all 8 batch rows
    const float* gi = gibuf + (size_t)(pbase + t) * BB * (3 * DD);
    int d = tid;
#pragma unroll
    for (int b = 0; b < BB; ++b) {
      size_t base = (size_t)b * (3 * DD);
      float r = sigmoidf_(gi[base + d] + gh[base + d]);
      float z = sigmoidf_(gi[base + DD + d] + gh[base + DD + d]);
      float n = tanhf(gi[base + 2 * DD + d] + r * gh[base + 2 * DD + d]);
      h[b][d] = (1.0f - z) * n + z * h[b][d];
    }
    __syncthreads();
  }

  // gate GEMM: [agg, h_cur] @ gate_w.T, K=2048 (LDS h | global x), N=1024
  const float* xc = x + ((size_t)mrow * SS + pos) * DD;
  for (int tile = wid; tile < DD / 16; tile += 32) {
    int nc = tile * 16 + lcol;
    const float* wrow = gate_w + (size_t)nc * (2 * DD);
    v8f acc = {0.f, 0.f, 0.f, 0.f, 0.f, 0.f, 0.f, 0.f};
#pragma unroll 4
    for (int kk = 0; kk < (2 * DD) / 4; ++kk) {
      int k0 = kk * 4 + 2 * half;
      bool lo = (k0 < DD);
      int kl = lo ? k0 : 0;          // clamped LDS index (both sides stay in range)
      int kg = lo ? 0 : (k0 - DD);   // clamped global index
      float a0 = lo ? h[mrow][kl] : xc[kg];
      float a1 = lo ? h[mrow][kl + 1] : xc[kg + 1];
      v2f a, b;
      a[0] = a0 * mmask;
      a[1] = a1 * mmask;
      b[0] = wrow[k0];
      b[1] = wrow[k0 + 1];
      acc = wmma4(a, b, acc);
    }
    if (half == 0) {
      float gb = gate_b[nc];
#pragma unroll
      for (int r = 0; r < 8; ++r) {
        float g = sigmoidf_(acc[r] + gb);
        float agg = h[r][nc];
        float hc = x[((size_t)r * SS + pos) * DD + nc];
        out[((size_t)r * SS + pos) * DD + nc] = g * agg + (1.0f - g) * hc;
      }
    }
  }
}

// ---------------------------------------------------------------------------
// Launch
// ---------------------------------------------------------------------------
extern "C" void kernel_launch(void* const* d_in, const int* in_sizes, int n_in,
                              void* d_out, int out_size, void* d_ws,
                              size_t ws_size, hipStream_t stream) {
  const float* x = (const float*)d_in[0];
  const float* pw_w1 = (const float*)d_in[1];
  const float* pw_b1 = (const float*)d_in[2];
  const float* pw_w2 = (const float*)d_in[3];
  const float* pw_b2 = (const float*)d_in[4];
  const float* msg_w = (const float*)d_in[5];
  const float* msg_b = (const float*)d_in[6];
  const float* ln_g = (const float*)d_in[7];
  const float* ln_b = (const float*)d_in[8];
  const float* gru_w_ih = (const float*)d_in[9];
  const float* gru_w_hh = (const float*)d_in[10];
  const float* gru_b_ih = (const float*)d_in[11];
  const float* gru_b_hh = (const float*)d_in[12];
  const float* gate_w = (const float*)d_in[13];
  const float* gate_b = (const float*)d_in[14];
  float* out = (float*)d_out;

  StructTable T;
  build_table(T);

  // 1) bulk copy (memory-bound part)
  int nvec = (BB * SS * DD) / 4;
  copy_kernel<<<nvec / 256, 256, 0, stream>>>((const f4*)x, (f4*)out, nvec);

  // workspace layout
  float* mbuf = (float*)d_ws;                               // npairs*B*D
  float* gibuf = mbuf + (size_t)T.npairs * BB * DD;         // npairs*B*3D
  float* ghbuf = gibuf + (size_t)T.npairs * BB * 3 * DD;    // nstruct*B*3D

  // 2) messages (LN + GELU + path weight), all pairs in parallel
  msg_kernel<<<T.npairs, 256, 0, stream>>>(x, pw_w1, pw_b1, pw_w2, pw_b2,
                                           msg_w, msg_b, ln_g, ln_b, mbuf, T);

  // 3) gi = m @ w_ih.T + b_ih, all pairs in parallel
  gi_kernel<<<dim3((unsigned)T.npairs, 24), 256, 0, stream>>>(mbuf, gru_w_ih,
                                                              gru_b_ih, gibuf);

  // 4) sequential GRU scan + gate + blend, one block per structure
  gru_gate_kernel<<<T.nstruct, 1024, 0, stream>>>(
      x, gibuf, gru_w_hh, gru_b_hh, gate_w, gate_b, ghbuf, out, T);
}